// FrustumProposerOG_29025388987095
// MI455X (gfx1250) — compile-verified
//
#include <hip/hip_runtime.h>
#include <math.h>

typedef __attribute__((ext_vector_type(2))) float v2f;
typedef __attribute__((ext_vector_type(8))) float v8f;

#define NDET   1024
#define MM     12
#define SS     4
#define RR     12
#define PP     (MM*SS*RR)   // 576
#define PI_F   3.14159265358979323846f

// template sign helpers (template row c, already *0.5)
__device__ __forceinline__ float tpl_x(int c) { return ((c >> 1) & 1) ? -0.5f : 0.5f; }
__device__ __forceinline__ float tpl_y(int c) { return (((c + 1) >> 1) & 1) ? -0.5f : 0.5f; }
__device__ __forceinline__ float tpl_z(int c) { return (c & 4) ? 0.5f : -0.5f; }

__device__ __forceinline__ float size_var(int s) {
    return 0.957f + (1.2f - 0.957f) * (float)s * (1.0f / 3.0f);
}

// ---------------------------------------------------------------------------
// Kernel 1: corner rotation table via V_WMMA_F32_16X16X4_F32.
// One wave per label. A(16x4) rows = (size_local, corner) -> [lx, lz, 0, 0],
// B(4x16) cols = rotation -> [cos, sin, 0, 0] (rx) / [-sin, cos, 0, 0] (rz).
// Tables: rxTab/rzTab [label][size][rot][corner], lyTab [label][size][corner].
// ---------------------------------------------------------------------------
__global__ __launch_bounds__(32) void corner_table_kernel(
    const float* __restrict__ anchors,
    float* __restrict__ rxTab, float* __restrict__ rzTab, float* __restrict__ lyTab) {
  const int l    = blockIdx.x;
  const int lane = threadIdx.x;
  const int col  = lane & 15;
  const bool lo  = lane < 16;

  const float aw = anchors[l * 3 + 0];
  const float ah = anchors[l * 3 + 1];
  const float al = anchors[l * 3 + 2];

  // B fragments: lanes 0-15 hold rows K=0,1 (valid); lanes 16-31 hold K=2,3 (zero pad)
  const int   rclamp = (col < 12) ? col : 11;          // cols 12..15 unused, keep finite
  const float ry = PI_F * (float)rclamp * (1.0f / 11.0f);
  const float cr = cosf(ry), sr = sinf(ry);
  v2f bRX, bRZ;
  bRX.x = lo ?  cr : 0.0f;  bRX.y = lo ? sr : 0.0f;
  bRZ.x = lo ? -sr : 0.0f;  bRZ.y = lo ? cr : 0.0f;

  #pragma unroll
  for (int t = 0; t < 2; ++t) {            // two row tiles: sizes {0,1} and {2,3}
    // A fragment: lanes 0-15 hold row M=col with K=0,1 = (lx, lz); lanes 16-31: K=2,3 = 0
    const int   sA = t * 2 + (col >> 3);
    const int   cA = col & 7;
    const float sv = size_var(sA);
    v2f a;
    a.x = lo ? aw * sv * tpl_x(cA) : 0.0f;
    a.y = lo ? al * sv * tpl_z(cA) : 0.0f;

    v8f acc = {};
    v8f dRX = __builtin_amdgcn_wmma_f32_16x16x4_f32(
        false, a, false, bRX, (short)0, acc, false, false);
    v8f dRZ = __builtin_amdgcn_wmma_f32_16x16x4_f32(
        false, a, false, bRZ, (short)0, acc, false, false);

    if (col < 12) {
      #pragma unroll
      for (int j = 0; j < 8; ++j) {
        const int M  = lo ? j : (j + 8);      // C/D layout: vgpr j -> M=j (lanes<16) / M=j+8
        const int s2 = t * 2 + (M >> 3);
        const int c2 = M & 7;
        const int o  = l * 384 + s2 * 96 + col * 8 + c2;
        rxTab[o] = dRX[j];
        rzTab[o] = dRZ[j];
      }
    }
  }

  // ly table (rotation-independent): 32 lanes = 4 sizes x 8 corners
  {
    const int s = lane >> 3, c = lane & 7;
    lyTab[l * 32 + lane] = ah * size_var(s) * tpl_y(c);
  }
}

// ---------------------------------------------------------------------------
// Kernel 2: per-detection proposal scoring + argmax. One wave32 per detection.
// ---------------------------------------------------------------------------
__global__ __launch_bounds__(256) void score_kernel(
    const float* __restrict__ boxes2d, const int* __restrict__ labels,
    const float* __restrict__ cam,     const float* __restrict__ anchors,
    const float* __restrict__ rxTab,   const float* __restrict__ rzTab,
    const float* __restrict__ lyTab,
    float* __restrict__ bscore, float* __restrict__ bproj, float* __restrict__ out) {
  const int n    = blockIdx.x * 8 + (threadIdx.x >> 5);
  const int lane = threadIdx.x & 31;

  __builtin_prefetch(&boxes2d[n * 4], 0, 1);   // global_prefetch_b8

  const float fx = cam[0], px = cam[2], fy = cam[4], py = cam[5];
  const float bx1 = boxes2d[n * 4 + 0], by1 = boxes2d[n * 4 + 1];
  const float bx2 = boxes2d[n * 4 + 2], by2 = boxes2d[n * 4 + 3];
  const int   lab = labels[n];

  const float cu = (bx1 + bx2) * 0.5f, cv = (by1 + by2) * 0.5f;
  const float r0 = (cu - px) / fx, r1 = (cv - py) / fy;
  const float nrm  = sqrtf(r0 * r0 + r1 * r1 + 1.0f);
  const float rayx = r0 / nrm, rayy = r1 / nrm, rayz = 1.0f / nrm;
  const float area_b = (bx2 - bx1) * (by2 - by1);

  const float* rxL = rxTab + lab * 384;
  const float* rzL = rzTab + lab * 384;
  const float* lyL = lyTab + lab * 32;

  float bs = -INFINITY; int bp = 0;
  float bu1 = 0.f, bv1 = 0.f, bu2 = 0.f, bv2 = 0.f;

  for (int k = 0; k < PP / 32; ++k) {          // 18 proposals per lane, ascending p
    const int p   = lane + (k << 5);
    const int im  = p / 48;
    const int rem = p - im * 48;
    const int isz = rem / 12;
    const int ir  = rem - isz * 12;

    const float mag = 2.0f + 48.0f * (float)im * (1.0f / 11.0f);
    const float cx = rayx * mag, cy = rayy * mag, cz = rayz * mag;

    const float* rxp = rxL + isz * 96 + ir * 8;
    const float* rzp = rzL + isz * 96 + ir * 8;
    const float* lyp = lyL + isz * 8;

    float umin = 1e30f, umax = -1e30f, vmin = 1e30f, vmax = -1e30f;
    #pragma unroll
    for (int c = 0; c < 8; ++c) {
      const float X  = cx + rxp[c];
      const float Yv = cy + lyp[c];
      const float Zv = fmaxf(cz + rzp[c], 0.1f);
      const float u  = fx * X  / Zv + px;
      const float v  = fy * Yv / Zv + py;
      umin = fminf(umin, u); umax = fmaxf(umax, u);
      vmin = fminf(vmin, v); vmax = fmaxf(vmax, v);
    }
    const float u1 = fminf(fmaxf(umin, 0.0f), 1600.0f);
    const float u2 = fminf(fmaxf(umax, 0.0f), 1600.0f);
    const float v1 = fminf(fmaxf(vmin, 0.0f), 900.0f);
    const float v2 = fminf(fmaxf(vmax, 0.0f), 900.0f);

    const float ix1 = fmaxf(u1, bx1), iy1 = fmaxf(v1, by1);
    const float ix2 = fminf(u2, bx2), iy2 = fminf(v2, by2);
    const float inter  = fmaxf(ix2 - ix1, 0.0f) * fmaxf(iy2 - iy1, 0.0f);
    const float area_p = fmaxf(u2 - u1, 0.0f) * fmaxf(v2 - v1, 0.0f);
    const float iou = inter / (area_p + area_b - inter + 1e-6f);
    const float sc  = 0.95f * iou + 0.226f * (1.0f - mag * (1.0f / 50.0f));

    if (sc > bs) { bs = sc; bp = p; bu1 = u1; bv1 = v1; bu2 = u2; bv2 = v2; }
  }

  // wave32 argmax reduction; ties -> lower proposal index (jnp.argmax semantics)
  #pragma unroll
  for (int off = 16; off > 0; off >>= 1) {
    float os  = __shfl_down(bs,  off, 32);
    int   op  = __shfl_down(bp,  off, 32);
    float ou1 = __shfl_down(bu1, off, 32);
    float ov1 = __shfl_down(bv1, off, 32);
    float ou2 = __shfl_down(bu2, off, 32);
    float ov2 = __shfl_down(bv2, off, 32);
    if (os > bs || (os == bs && op < bp)) {
      bs = os; bp = op; bu1 = ou1; bv1 = ov1; bu2 = ou2; bv2 = ov2;
    }
  }

  if (lane == 0) {
    const int im  = bp / 48;
    const int rem = bp - im * 48;
    const int isz = rem / 12;
    const int ir  = rem - isz * 12;
    const float mag = 2.0f + 48.0f * (float)im * (1.0f / 11.0f);
    const float sv  = size_var(isz);
    out[n * 8 + 0] = rayx * mag;
    out[n * 8 + 1] = rayy * mag;
    out[n * 8 + 2] = rayz * mag;
    out[n * 8 + 3] = anchors[lab * 3 + 0] * sv;
    out[n * 8 + 4] = anchors[lab * 3 + 1] * sv;
    out[n * 8 + 5] = anchors[lab * 3 + 2] * sv;
    out[n * 8 + 6] = PI_F * (float)ir * (1.0f / 11.0f);
    bscore[n] = bs;
    bproj[n * 4 + 0] = bu1; bproj[n * 4 + 1] = bv1;
    bproj[n * 4 + 2] = bu2; bproj[n * 4 + 3] = bv2;
  }
}

// ---------------------------------------------------------------------------
// Kernel 3: stable descending bitonic argsort + greedy NMS, single block in LDS.
// ---------------------------------------------------------------------------
__global__ __launch_bounds__(1024) void nms_kernel(
    const float* __restrict__ bscore, const float* __restrict__ bproj,
    float* __restrict__ out) {
  __shared__ float sk[NDET];
  __shared__ int   si[NDET];
  __shared__ float sx1[NDET], sy1[NDET], sx2[NDET], sy2[NDET], sar[NDET];
  __shared__ int   skeep[NDET];

  const int tid = threadIdx.x;
  sk[tid] = bscore[tid];
  si[tid] = tid;
  __syncthreads();

  // bitonic sort: rank order = (score desc, idx asc) -> emulates stable argsort(-score)
  for (int k = 2; k <= NDET; k <<= 1) {
    for (int j = k >> 1; j > 0; j >>= 1) {
      const int ixj = tid ^ j;
      if (ixj > tid) {
        const float s1 = sk[tid], s2 = sk[ixj];
        const int   i1 = si[tid], i2 = si[ixj];
        const bool before = (s1 > s2) || (s1 == s2 && i1 < i2); // tid elem ranks first
        const bool asc = ((tid & k) == 0);
        if (asc ? !before : before) {
          sk[tid] = s2; si[tid] = i2;
          sk[ixj] = s1; si[ixj] = i1;
        }
      }
      __syncthreads();
    }
  }

  const int   oi = si[tid];
  const float x1 = bproj[oi * 4 + 0], y1 = bproj[oi * 4 + 1];
  const float x2 = bproj[oi * 4 + 2], y2 = bproj[oi * 4 + 3];
  const float ar = fmaxf(x2 - x1, 0.0f) * fmaxf(y2 - y1, 0.0f);
  sx1[tid] = x1; sy1[tid] = y1; sx2[tid] = x2; sy2[tid] = y2;
  sar[tid] = ar;
  skeep[tid] = 1;

  for (int i = 0; i < NDET; ++i) {
    __syncthreads();
    if (tid > i && skeep[i]) {
      const float nx1 = fmaxf(x1, sx1[i]);
      const float ny1 = fmaxf(y1, sy1[i]);
      const float nx2 = fminf(x2, sx2[i]);
      const float ny2 = fminf(y2, sy2[i]);
      const float inter = fmaxf(nx2 - nx1, 0.0f) * fmaxf(ny2 - ny1, 0.0f);
      const float iou = inter / (ar + sar[i] - inter + 1e-6f);
      if (iou > 0.7f) skeep[tid] = 0;
    }
  }
  __syncthreads();

  const float s = sk[tid];
  out[oi * 8 + 7] = (skeep[tid] && s > 0.1f) ? s : 0.0f;
}

// ---------------------------------------------------------------------------
extern "C" void kernel_launch(void* const* d_in, const int* in_sizes, int n_in,
                              void* d_out, int out_size, void* d_ws, size_t ws_size,
                              hipStream_t stream) {
  const float* boxes2d = (const float*)d_in[0];
  const int*   labels  = (const int*)d_in[1];
  const float* cam     = (const float*)d_in[2];
  const float* anchors = (const float*)d_in[3];

  float* ws     = (float*)d_ws;
  float* rxTab  = ws;            // 10*4*12*8 = 3840
  float* rzTab  = ws + 3840;     // 3840
  float* lyTab  = ws + 7680;     // 10*4*8 = 320
  float* bscore = ws + 8000;     // 1024
  float* bproj  = ws + 9024;     // 4096
  float* out    = (float*)d_out; // 1024*8

  corner_table_kernel<<<10, 32, 0, stream>>>(anchors, rxTab, rzTab, lyTab);
  score_kernel<<<NDET / 8, 256, 0, stream>>>(boxes2d, labels, cam, anchors,
                                             rxTab, rzTab, lyTab, bscore, bproj, out);
  nms_kernel<<<1, NDET, 0, stream>>>(bscore, bproj, out);
}